// HashGridEncoder2D_58703613002181
// MI455X (gfx1250) — compile-verified
//
#include <hip/hip_runtime.h>
#include <cstdint>
#include <cstddef>

// ---------------------------------------------------------------------------
// HashGridEncoder2D for MI455X (gfx1250)
//
// Strategy (memory-bound kernel, WMMA not applicable):
//  * grid = (query_chunks x 12 levels); each workgroup handles one level's
//    table (exactly 256 KB = 32768 x float2) for a chunk of 8192 queries.
//  * The level table is DMA'd into LDS with a single TENSOR_LOAD_TO_LDS
//    (TDM, TENSORcnt) -> all 4-corner random gathers become ds_load_b64.
//  * coords read coalesced (float2/lane), output written as float2 per
//    (query, level) slice.
// ---------------------------------------------------------------------------

#define NUM_LEVELS 12
#define FEATS      2
#define LOG2_HASH  15
#define HASH_SIZE  (1 << LOG2_HASH)
#define HASH_MASK  (HASH_SIZE - 1)
#define TABLE_BYTES (HASH_SIZE * FEATS * 4)   // 262144 bytes = 256 KB (fits 320 KB LDS)

typedef __attribute__((ext_vector_type(4))) unsigned int u32x4;
typedef __attribute__((ext_vector_type(8))) int          i32x8;
typedef __attribute__((ext_vector_type(4))) int          i32x4;

__device__ __forceinline__ int hash2(int ix, int iy) {
    // low 15 bits identical under int32 wraparound; do it in uint32 (no UB)
    unsigned h = (unsigned)ix * 73856093u ^ (unsigned)iy * 19349663u;
    return (int)(h & HASH_MASK);
}

template <int THREADS, int QPT>
__global__ __launch_bounds__(THREADS)
void hashgrid2d_kernel(const float2* __restrict__ coords,
                       const float*  __restrict__ tables,
                       float*        __restrict__ out,
                       int nQueries)
{
    extern __shared__ float2 smem[];   // HASH_SIZE float2 entries (256 KB dynamic LDS)

    const int level = (int)(blockIdx.x) % NUM_LEVELS;
    const int chunk = (int)(blockIdx.x) / NUM_LEVELS;
    const int base  = chunk * (THREADS * QPT);

    // ---- Stage this level's 256 KB table into LDS via the Tensor Data Mover.
    // TDM ignores EXEC but issues once per executing wave, so gate to a single
    // wave (threadIdx.x == 0 -> only wave 0 reaches the instruction).
    if (threadIdx.x == 0) {
        const unsigned long long gaddr =
            (unsigned long long)(uintptr_t)tables +
            (unsigned long long)level * (unsigned long long)TABLE_BYTES;

        // D# group 0: count=1 | lds_addr | global_addr[56:0] | type=2
        u32x4 g0 = { 1u,                                  // count=1, user desc
                     0u,                                  // lds_addr = 0 (dynamic LDS base)
                     (unsigned)(gaddr & 0xFFFFFFFFull),   // global_addr[31:0]
                     (unsigned)((gaddr >> 32) & 0x1FFFFFFull) | (2u << 30) }; // [56:32] | type=2

        // D# group 1: 1-row tile of 32768 x 8-byte elements (whole table)
        i32x8 g1 = { (int)(3u << 16),        // workgroup_mask=0, data_size=3 (8 bytes)
                     (int)(0x8000u << 16),   // tensor_dim0 = 32768 (lo16 -> w1[31:16])
                     (int)(1u << 16),        // tensor_dim0 hi=0 ; tensor_dim1 = 1
                     (int)(0x8000u << 16),   // tensor_dim1 hi=0 ; tile_dim0 = 32768
                     1,                      // tile_dim1 = 1 ; tile_dim2 = 0
                     32768,                  // tensor_dim0_stride (elements)
                     0,                      // stride hi / tensor_dim1_stride lo
                     0 };
        i32x4 gz4 = { 0, 0, 0, 0 };          // groups 2/3 unused (<=2D tensor)
        i32x8 gz8 = { 0, 0, 0, 0, 0, 0, 0, 0 };

        // clang-23 / amdgpu-toolchain 6-arg form:
        // (g0, g1, g2, g3, extra_group, cpol)
        __builtin_amdgcn_tensor_load_to_lds(g0, g1, gz4, gz4, gz8, 0);
        __builtin_amdgcn_s_wait_tensorcnt(0);   // DMA complete before releasing barrier
    }
    __syncthreads();

    const float res_f = (float)((16 << level) - 1);   // res-1, res = 16 * 2^level

#pragma unroll
    for (int k = 0; k < QPT; ++k) {
        const int q = base + (int)threadIdx.x + k * THREADS;
        if (q < nQueries) {
            const float2 c = coords[q];                 // coalesced global_load_b64
            const float x0 = (c.x + 1.0f) * 0.5f;
            const float x1 = (c.y + 1.0f) * 0.5f;
            // NOTE: reference swaps axes: gx from coord[...,1], gy from coord[...,0]
            const float gx = x1 * res_f;
            const float gy = x0 * res_f;

            const float ix0f = fminf(fmaxf(floorf(gx), 0.0f), res_f);
            const float iy0f = fminf(fmaxf(floorf(gy), 0.0f), res_f);
            const float fx = gx - ix0f;
            const float fy = gy - iy0f;
            const int ix0 = (int)ix0f;
            const int iy0 = (int)iy0f;
            const int ix1 = ix0 + 1;
            const int iy1 = iy0 + 1;

            // 4 random gathers out of LDS (ds_load_b64 each)
            const float2 e00 = smem[hash2(ix0, iy0)];
            const float2 e10 = smem[hash2(ix1, iy0)];
            const float2 e01 = smem[hash2(ix0, iy1)];
            const float2 e11 = smem[hash2(ix1, iy1)];

            const float w00 = (1.0f - fx) * (1.0f - fy);
            const float w10 = fx * (1.0f - fy);
            const float w01 = (1.0f - fx) * fy;
            const float w11 = fx * fy;

            float2 feat;
            feat.x = e00.x * w00 + e10.x * w10 + e01.x * w01 + e11.x * w11;
            feat.y = e00.y * w00 + e10.y * w10 + e01.y * w01 + e11.y * w11;

            // out is (B*Q, L*F); this level's 8-byte slice (8-byte aligned)
            *(float2*)(out + (size_t)q * (NUM_LEVELS * FEATS) + level * FEATS) = feat;
        }
    }
}

extern "C" void kernel_launch(void* const* d_in, const int* in_sizes, int n_in,
                              void* d_out, int out_size, void* d_ws, size_t ws_size,
                              hipStream_t stream) {
    (void)n_in; (void)out_size; (void)d_ws; (void)ws_size;

    const float2* coords = (const float2*)d_in[0];   // (4, 262144, 2) f32
    const float*  tables = (const float*)d_in[1];    // (12, 32768, 2) f32
    float*        out    = (float*)d_out;            // (4, 262144, 24) f32

    const int nQueries = in_sizes[0] / 2;            // B*Q = 1,048,576

    constexpr int THREADS = 512;   // 16 waves (wave32) per workgroup
    constexpr int QPT     = 16;    // 8192 queries per workgroup chunk
    const int perChunk = THREADS * QPT;
    const int chunks   = (nQueries + perChunk - 1) / perChunk;   // 128

    // 256 KB dynamic LDS per workgroup (gfx1250 WGP has 320 KB)
    (void)hipFuncSetAttribute((const void*)hashgrid2d_kernel<THREADS, QPT>,
                              hipFuncAttributeMaxDynamicSharedMemorySize, TABLE_BYTES);

    dim3 grid(chunks * NUM_LEVELS);   // 1536 workgroups, one (chunk, level) each
    hashgrid2d_kernel<THREADS, QPT><<<grid, THREADS, TABLE_BYTES, stream>>>(
        coords, tables, out, nQueries);
}